// WaveletTransform_34754875359595
// MI455X (gfx1250) — compile-verified
//
#include <hip/hip_runtime.h>

// ---------------------------------------------------------------------------
// Haar DWT2 (pywt 'haar', single level) for x[8,64,512,512] f32.
// Memory-bound (AI ~0.5 flop/byte): 1 GiB streaming traffic / 23.3 TB/s
// => ~46us floor. WMMA is deliberately NOT used (it cannot move a bandwidth
// bound). CDNA5 path: async global->LDS B128 staging (ASYNCcnt-tracked,
// 2-deep double-buffered pipeline) + non-temporal B64 stores (data set is
// 1 GiB >> 192 MB L2, so stream past the cache).
//
// Addressing identity: input offset for work item idx is
//   (idx>>7)*1024 + (idx&127)*4   (floats), bottom row at +512
// and the per-plane output offset is exactly 2*idx. With grid stride T
// (T % 128 == 0) both advance by constants (8T floats / 2T elems), so the
// steady-state loop is pointer-bump only.
// ---------------------------------------------------------------------------

typedef float f32x4 __attribute__((ext_vector_type(4)));
typedef float f32x2 __attribute__((ext_vector_type(2)));
typedef int   v4i   __attribute__((ext_vector_type(4)));

#define AS1 __attribute__((address_space(1)))
#define AS3 __attribute__((address_space(3)))

constexpr int TPB   = 256;  // 8 wave32 per block
constexpr int ITERS = 4;    // work items per thread (pipelined)

#if __has_builtin(__builtin_amdgcn_global_load_async_to_lds_b128)
#define HAVE_ASYNC_BUILTIN 1
#else
#define HAVE_ASYNC_BUILTIN 0
#endif

__device__ __forceinline__ void async_copy16(const float* g, f32x4* lds) {
#if HAVE_ASYNC_BUILTIN
  // Probe-confirmed signature: (v4i AS1* gsrc, v4i AS3* lds_dst, Ii off, Ii cpol)
  v4i* gg = (v4i*)g;
  v4i* ll = (v4i*)lds;
  __builtin_amdgcn_global_load_async_to_lds_b128(
      (AS1 v4i*)gg, (AS3 v4i*)ll, /*offset=*/0, /*cpol=*/0);
#else
  asm volatile("global_load_async_to_lds_b128 %0, %1, off"
               :: "v"((unsigned)(unsigned long long)(AS3 char*)(void*)lds),
                  "v"(g)
               : "memory");
#endif
}

#if __has_builtin(__builtin_amdgcn_s_wait_asynccnt)
#define WAIT_ASYNC(n) __builtin_amdgcn_s_wait_asynccnt(n)
#else
#define WAIT_ASYNC(n) asm volatile("s_wait_asynccnt " #n ::: "memory")
#endif

// Butterfly + 4 NT plane stores for one (top,bot) float4 pair.
__device__ __forceinline__ void haar_emit(const f32x4 t, const f32x4 u,
                                          float* __restrict__ out,
                                          const long long N, const long long o) {
  const float s0 = t.x + u.x, m0 = t.x - u.x;
  const float s1 = t.y + u.y, m1 = t.y - u.y;
  const float s2 = t.z + u.z, m2 = t.z - u.z;
  const float s3 = t.w + u.w, m3 = t.w - u.w;
  const f32x2 ll = { 0.5f * (s0 + s1), 0.5f * (s2 + s3) };  // a+b+c+d
  const f32x2 lh = { 0.5f * (m0 + m1), 0.5f * (m2 + m3) };  // a+b-c-d
  const f32x2 hl = { 0.5f * (s0 - s1), 0.5f * (s2 - s3) };  // a-b+c-d
  const f32x2 hh = { 0.5f * (m0 - m1), 0.5f * (m2 - m3) };  // a-b-c+d
  __builtin_nontemporal_store(ll, (f32x2*)(out + o));
  __builtin_nontemporal_store(lh, (f32x2*)(out + N + o));
  __builtin_nontemporal_store(hl, (f32x2*)(out + 2 * N + o));
  __builtin_nontemporal_store(hh, (f32x2*)(out + 3 * N + o));
}

__global__ __launch_bounds__(TPB) void haar_dwt2_kernel(
    const float* __restrict__ x, float* __restrict__ out,
    const long long N /*per-plane out elems*/, const long long P /*work items*/)
{
  __shared__ f32x4 sh[2][2][TPB];   // [stage][top/bot][lane] = 16 KB
  const int tid = threadIdx.x;
  const long long T  = (long long)gridDim.x * TPB;      // grid stride, %128==0
  const long long g0 = (long long)blockIdx.x * TPB + tid;

  if (g0 + (long long)(ITERS - 1) * T < P) {
    // ---- fast path: all ITERS stages in range, branch-free pipeline ----
    const long long DIN  = T << 3;   // input advance per stage (floats)
    const long long DOUT = T << 1;   // output advance per stage (elems)
    const float* gt = x + ((g0 >> 7) << 10) + ((g0 & 127) << 2);
    const float* gb = gt + 512;
    long long o = g0 << 1;

    // Pipeline prologue: stage 0 in flight.
    async_copy16(gt, &sh[0][0][tid]);
    async_copy16(gb, &sh[0][1][tid]);

#pragma unroll
    for (int k = 0; k < ITERS; ++k) {
      const int cur = k & 1;
      const float* gtn = gt + DIN;
      const float* gbn = gb + DIN;
      if (k + 1 < ITERS) {
        async_copy16(gtn, &sh[(k + 1) & 1][0][tid]);
        async_copy16(gbn, &sh[(k + 1) & 1][1][tid]);
        WAIT_ASYNC(2);   // stage k's 2 loads complete; stage k+1 in flight
      } else {
        WAIT_ASYNC(0);   // drain last stage
      }
      asm volatile("" ::: "memory");

      const f32x4 t = sh[cur][0][tid];  // a0 b0 a1 b1 (input row 2*ho)
      const f32x4 u = sh[cur][1][tid];  // c0 d0 c1 d1 (input row 2*ho+1)
      haar_emit(t, u, out, N, o);

      gt = gtn; gb = gbn; o += DOUT;
    }
  } else {
    // ---- tail path (rare): guarded, direct loads, no pipelining ----
    for (int k = 0; k < ITERS; ++k) {
      const long long idx = g0 + (long long)k * T;
      if (idx < P) {
        const float* p = x + ((idx >> 7) << 10) + ((idx & 127) << 2);
        const f32x4 t = *(const f32x4*)p;
        const f32x4 u = *(const f32x4*)(p + 512);
        haar_emit(t, u, out, N, idx << 1);
      }
    }
  }
}

extern "C" void kernel_launch(void* const* d_in, const int* in_sizes, int n_in,
                              void* d_out, int out_size, void* d_ws, size_t ws_size,
                              hipStream_t stream) {
  (void)n_in; (void)out_size; (void)d_ws; (void)ws_size;
  const float* x   = (const float*)d_in[0];
  float*       out = (float*)d_out;
  const long long in_total = (long long)in_sizes[0];   // B*C*H*W
  const long long N = in_total / 4;                    // per-plane outputs
  const long long P = in_total / 8;                    // work items (2 px each)
  const long long per_blk = (long long)TPB * ITERS;
  const int blocks = (int)((P + per_blk - 1) / per_blk);  // 16384 for ref shape
  haar_dwt2_kernel<<<blocks, TPB, 0, stream>>>(x, out, N, P);
}